// GraphUpSample_Conv_12120397709986
// MI455X (gfx1250) — compile-verified
//
#include <hip/hip_runtime.h>

// ---------------------------------------------------------------------------
// GraphUpSample_Conv for MI455X (gfx1250, wave32, WMMA).
//
// Folded math:
//   Weff[n,i,c,k] = sum_o W1[n,i,o,k] * W2[n,o,c]          (32 x 16 x 16 x 2)
//   beff[n,c]     = sum_o b1[n,o]*W2[n,o,c] + b2[n,c]
//   out[b,c,f,2n+k] = sum_i x[b,i,f,n] * Weff[n,i,c,k] + beff[n,c]
// (INDEX permutation in the reference is the identity.)
//
// Memory-bound: 128MB in + 256MB out => ~16.5us floor @ 23.3 TB/s.  All
// global traffic is 128B-coalesced; operand shuffling happens in LDS.
// Weights are pre-folded into V_WMMA_F32_16X16X4_F32 B-fragment order so the
// inner loop loads each B fragment with one coalesced b64.
// ---------------------------------------------------------------------------

typedef __attribute__((ext_vector_type(2))) float v2f;
typedef __attribute__((ext_vector_type(8))) float v8f;

#define B_    128
#define CIN   16
#define COUT  16
#define F_    512
#define N_    32
#define K_    2

// d_ws layout (floats):
//   Wfrag[n][512] : fragment-ordered folded weights, 32*512 = 16384 floats
//       element (n, h, q, half, c, e) at
//       n*512 + (((h*4 + q)*2 + half)*16 + c)*2 + e ,  i = 4q + 2*half + e
//   beff[n][c]    : 512 floats
#define WF_ELEMS (N_ * K_ * CIN * COUT)

// xs LDS layout: xs[i*XS_ISTR + n*17 + row]
//  - n-stride 17 (odd)      -> staging stores conflict-free across lanes
//  - i-stride 552 = 32*17+8 -> A-fragment reads: half-wave bases differ by
//    2*552 = 1104 = 16 (mod 64), so lanes 0-15 (banks b..b+15) and lanes
//    16-31 (banks b+16..b+31) hit disjoint bank groups: conflict-free.
#define XS_ISTR 552
#define XS_SIZE (16 * XS_ISTR)   // 8832 floats = 34.5 KB

// ------------------------------ precompute ---------------------------------
__global__ __launch_bounds__(256) void precompute_eff(
    const float* __restrict__ W1, const float* __restrict__ b1,
    const float* __restrict__ W2, const float* __restrict__ b2,
    float* __restrict__ Wfrag, float* __restrict__ beff) {
  const int n = blockIdx.x;          // one block per node
  const int t = threadIdx.x;
  const float* W1n = W1 + (size_t)n * CIN * COUT * K_;   // [i][o][k]
  const float* W2n = W2 + (size_t)n * COUT * COUT;       // [o][c]

  for (int e = t; e < K_ * CIN * COUT; e += 256) {       // 512 fragment elems
    const int el = e & 1;            // pair element
    const int c  = (e >> 1) & 15;
    const int hf = (e >> 5) & 1;     // half-wave
    const int q  = (e >> 6) & 3;     // K-chunk
    const int h  = (e >> 8) & 1;     // k
    const int i  = 4 * q + 2 * hf + el;
    float s = 0.f;
    for (int o = 0; o < COUT; ++o)
      s += W1n[(i * COUT + o) * K_ + h] * W2n[o * COUT + c];
    Wfrag[(size_t)n * 512 + e] = s;
  }
  if (t < COUT) {
    float s = b2[n * COUT + t];
    for (int o = 0; o < COUT; ++o)
      s += b1[n * COUT + o] * W2n[o * COUT + t];
    beff[n * COUT + t] = s;
  }
}

// ------------------------------ main kernel --------------------------------
// Block = 256 threads = 8 wave32s, one 16-row (b,f) tile, all 32 nodes.
// Wave w handles nodes 4w..4w+3.  Per node: D(16x32) = A(16x16) * Weff(16x32)
// via 4x V_WMMA_F32_16X16X4_F32 per 16-column half, bias preloaded in C.
__global__ __launch_bounds__(256, 1) void upsample_conv_wmma(
    const float* __restrict__ x, const float* __restrict__ Wfrag,
    const float* __restrict__ beff, float* __restrict__ out) {
  // Input staging (transpose x[b,i,f,n] -> per-node column-major A tiles (row fastest)).
  __shared__ __align__(16) float xs[XS_SIZE];
  // Output staging: logical [row(16)][c(16)][m(64)] f32 = exactly 64KB.
  // Physical word index = row*1024 + c*64 + (m ^ (4*c))  -> conflict-free
  // D-writes (distinct banks across c) and conflict-free float4 readback.
  __shared__ __align__(16) float ys[16 * 16 * 64];

  const int t    = threadIdx.x;
  const int lane = t & 31;
  const int wave = t >> 5;           // 0..7
  const int half = lane >> 4;        // 0..1
  const int lc   = lane & 15;        // row (for A/D) or column c (for B/D)

  const int tile = blockIdx.x;       // 0..4095
  const int row0 = tile * 16;        // global (b,f) row base; rows never cross b
  const int b    = row0 >> 9;        // / F_
  const int f0   = row0 & 511;       // tile-aligned, f0+15 < 512

  // ---- stage the 16x16x32 x-tile into LDS, fully coalesced b128 loads ----
  // thread t: n4 = t&7 (float4 over n), pr = t>>3; 8 passes cover all
  // 256 (row,i) runs.  Threads 0..7 read one contiguous 128B run.
  {
    const int n4 = t & 7;
    const int pr = t >> 3;
#pragma unroll
    for (int j = 0; j < 8; ++j) {
      const int pair = j * 32 + pr;  // 0..255
      const int row  = pair >> 4;
      const int i    = pair & 15;
      const float4 v = *(const float4*)&x[
          (((size_t)b * CIN + i) * F_ + (f0 + row)) * N_ + 4 * n4];
      const int base = i * XS_ISTR + row;
      xs[base + (4 * n4 + 0) * 17] = v.x;
      xs[base + (4 * n4 + 1) * 17] = v.y;
      xs[base + (4 * n4 + 2) * 17] = v.z;
      xs[base + (4 * n4 + 3) * 17] = v.w;
    }
  }
  __syncthreads();

  // ------------------------------ WMMA compute -----------------------------
  for (int nn = 0; nn < 4; ++nn) {
    const int n = wave * 4 + nn;
    const float bias = beff[n * COUT + lc];
    v8f acc0, acc1;
#pragma unroll
    for (int r = 0; r < 8; ++r) { acc0[r] = bias; acc1[r] = bias; }

    // B fragments in fragment order: one coalesced b64 per (h,q).
    const v2f* wfr = (const v2f*)(Wfrag + (size_t)n * 512);

#pragma unroll
    for (int q = 0; q < 4; ++q) {
      const int i0 = 4 * q + 2 * half;   // K-lane mapping of 16x4 f32 A / 4x16 B
      v2f a;
      // A[row=lc, K=i0], A[row=lc, K=i0+1] from LDS (conflict-free)
      a[0] = xs[i0 * XS_ISTR + n * 17 + lc];
      a[1] = xs[(i0 + 1) * XS_ISTR + n * 17 + lc];
      // B[K, N=lc] fragment pairs for h=0 / h=1 (single b64 each, L2-hot)
      const v2f bh0 = wfr[((0 * 4 + q) * 2 + half) * 16 + lc];
      const v2f bh1 = wfr[((1 * 4 + q) * 2 + half) * 16 + lc];

      acc0 = __builtin_amdgcn_wmma_f32_16x16x4_f32(
          false, a, false, bh0, (short)0, acc0, false, false);
      acc1 = __builtin_amdgcn_wmma_f32_16x16x4_f32(
          false, a, false, bh1, (short)0, acc1, false, false);
    }

    // D layout: vgpr r, lane -> row = r + 8*half, col c = lc.
    // Stage into swizzled LDS; (2n ^ 4c) is even so the pair is contiguous.
    const int sw = 4 * lc;
    const int m0 = (2 * n) ^ sw;
#pragma unroll
    for (int r = 0; r < 8; ++r) {
      const int row = r + 8 * half;
      float2 p; p.x = acc0[r]; p.y = acc1[r];
      *(float2*)&ys[row * 1024 + lc * 64 + m0] = p;
    }
  }
  __syncthreads();

  // Coalesced store: iteration j stores row j; 16 threads cover one 256B run.
  //   thread t: c = t>>4, m4 = t&15  ->  64 consecutive floats per 16 lanes.
  const int c  = t >> 4;
  const int m4 = t & 15;
  const int sw = 4 * c;
  const size_t obase = (((size_t)b * COUT + c) * F_ + f0) * (N_ * K_) + 4 * m4;
#pragma unroll 4
  for (int j = 0; j < 16; ++j) {
    const float4 v = *(const float4*)&ys[j * 1024 + c * 64 + ((4 * m4) ^ sw)];
    *(float4*)(out + obase + (size_t)j * (N_ * K_)) = v;
  }
}

// ------------------------------ launch -------------------------------------
extern "C" void kernel_launch(void* const* d_in, const int* in_sizes, int n_in,
                              void* d_out, int out_size, void* d_ws, size_t ws_size,
                              hipStream_t stream) {
  (void)in_sizes; (void)n_in; (void)out_size; (void)ws_size;
  const float* x  = (const float*)d_in[0];
  const float* W1 = (const float*)d_in[1];
  const float* b1 = (const float*)d_in[2];
  const float* W2 = (const float*)d_in[3];
  const float* b2 = (const float*)d_in[4];
  float* outp  = (float*)d_out;
  float* Wfrag = (float*)d_ws;
  float* beff  = Wfrag + WF_ELEMS;

  precompute_eff<<<N_, 256, 0, stream>>>(W1, b1, W2, b2, Wfrag, beff);

  const int tiles = (B_ * F_) / 16;  // 4096
  upsample_conv_wmma<<<tiles, 256, 0, stream>>>(x, Wfrag, beff, outp);
}